// SigmaDockDenoiser_84842783965756
// MI455X (gfx1250) — compile-verified
//
#include <hip/hip_runtime.h>
#include <hip/hip_bf16.h>

#define BG 8192      // graphs
#define AG 1024      // atoms per graph
#define FG 32        // max fragments per graph
#define VIRT 4       // ligand_virtual entity index

typedef __attribute__((ext_vector_type(2))) float v2f;
typedef __attribute__((ext_vector_type(8))) float v8f;
typedef __attribute__((ext_vector_type(4))) float vf4;
typedef __attribute__((ext_vector_type(4))) int   vi4;

// ---------------------------------------------------------------------------
// Kernel A: per-graph max fragment index -> nfrag[b] = fmax + 1
// One block (256 thr) per graph; graph atoms are contiguous (batch_idx=i/AG).
// ---------------------------------------------------------------------------
__global__ void __launch_bounds__(256) fmax_kernel(const int* __restrict__ frag,
                                                   int* __restrict__ nfrag) {
    int b = blockIdx.x, t = threadIdx.x;
    vi4 f = *(const vi4*)(frag + (size_t)b * AG + (size_t)t * 4);
    int m = max(max(f.x, f.y), max(f.z, f.w));
    __shared__ int sm[256];
    sm[t] = m;
    __syncthreads();
    for (int s = 128; s > 0; s >>= 1) {
        if (t < s) sm[t] = max(sm[t], sm[t + s]);
        __syncthreads();
    }
    if (t == 0) nfrag[b] = sm[0] + 1;   // frag >= -1 always, so fmax+1 == max(fmax,-1)+1
}

// ---------------------------------------------------------------------------
// Kernel B: exclusive scan over nfrag[8192] -> offset[8192]. Single block.
// 256 threads x 32 contiguous graphs each; Hillis-Steele over thread totals.
// ---------------------------------------------------------------------------
__global__ void __launch_bounds__(256) scan_kernel(const int* __restrict__ nfrag,
                                                   int* __restrict__ offs) {
    __shared__ int ss[256];
    int t = threadIdx.x;
    int g0 = t * 32;
    int loc[32];
    int tot = 0;
    #pragma unroll
    for (int i = 0; i < 32; ++i) { loc[i] = tot; tot += nfrag[g0 + i]; }
    ss[t] = tot;
    __syncthreads();
    for (int d = 1; d < 256; d <<= 1) {
        int v = (t >= d) ? ss[t - d] : 0;
        __syncthreads();
        ss[t] += v;
        __syncthreads();
    }
    int base = ss[t] - tot;   // exclusive prefix of this thread's chunk
    #pragma unroll
    for (int i = 0; i < 32; ++i) offs[g0 + i] = base + loc[i];
}

// ---------------------------------------------------------------------------
// Kernel C: zero the coms+cnt region (BG*FG*4 floats = 4 MB).
// ---------------------------------------------------------------------------
__global__ void __launch_bounds__(256) zero_kernel(float* __restrict__ out) {
    size_t i = (size_t)blockIdx.x * 256 + threadIdx.x;   // 1024*256 = BG*FG
    vf4 z = {0.f, 0.f, 0.f, 0.f};
    *(vf4*)(out + i * 4) = z;
}

// ---------------------------------------------------------------------------
// Kernel D: fused flat-index + COM. One block (256 thr = 8 waves) per graph.
//  - pos tile (12 KB) staged to LDS with async global->LDS loads
//  - per-wave segment-sum via V_WMMA_F32_16X16X4_F32 one-hot matmuls
//  - cross-wave merge via ds_add_f32 into 32x4 LDS bins, then final write
// ---------------------------------------------------------------------------
__global__ void __launch_bounds__(256) com_kernel(
    const float* __restrict__ pos, const int* __restrict__ frag,
    const int* __restrict__ ent, const int* __restrict__ msk,
    const int* __restrict__ nfrag, const int* __restrict__ offs,
    float* __restrict__ out) {
    __shared__ float sp[AG * 3];     // staged positions (x,y,z per atom)
    __shared__ int   q[AG];          // valid ? frag : -1
    __shared__ float bins[FG * 4];   // [bin][x,y,z,cnt]

    int b = blockIdx.x, t = threadIdx.x;
    size_t gbase = (size_t)b * AG;
    int offb = offs[b];
    int nfr  = nfrag[b];

    // ---- async stage of pos[gbase*3 .. +3072) into LDS (3 x 4KB passes) ----
    unsigned spAddr = (unsigned)(unsigned long long)(void*)sp;  // flat->LDS: low 32 bits
    const float* gp = pos + gbase * 3;                          // uniform -> SGPR pair
    #pragma unroll
    for (int p = 0; p < 3; ++p) {
        unsigned off = (unsigned)t * 16u + (unsigned)p * 4096u;
        asm volatile("global_load_async_to_lds_b128 %0, %1, %2 offset:0"
                     :: "v"(spAddr + off), "v"(off), "s"(gp)
                     : "memory");
    }

    // ---- phase 1: flat index output + validity staging ----
    vi4 fr = *(const vi4*)(frag + gbase + (size_t)t * 4);
    vi4 en = *(const vi4*)(ent  + gbase + (size_t)t * 4);
    vi4 mk = *(const vi4*)(msk  + gbase + (size_t)t * 4);
    int fv[4] = {fr.x, fr.y, fr.z, fr.w};
    int ev[4] = {en.x, en.y, en.z, en.w};
    int mv[4] = {mk.x, mk.y, mk.z, mk.w};
    vf4 flat;
    #pragma unroll
    for (int i = 0; i < 4; ++i) {
        bool lig   = fv[i] >= 0;
        bool valid = lig && (mv[i] != 0) && (ev[i] != VIRT);
        flat[i]    = lig ? (float)(fv[i] + offb) : -1.0f;
        q[t * 4 + i] = valid ? fv[i] : -1;
    }
    *(vf4*)(out + (size_t)BG * FG * 4 + gbase + (size_t)t * 4) = flat;
    if (t < FG * 4) bins[t] = 0.0f;

    asm volatile("s_wait_asynccnt 0" ::: "memory");
    __syncthreads();

    // ---- phase 2: one-hot WMMA accumulation (uniform control flow) ----
    int lane = t & 31;
    int wv   = t >> 5;
    int hh   = lane >> 4;            // half select: K-pair 0/1 vs 2/3
    int lm   = lane & 15;            // A: bin row (0..15); B/D: column
    int chc  = lm < 3 ? lm : 0;      // clamped channel -> safe LDS address, all lanes
    // Loop-invariant B-column shaping: col<3 -> pos channel, col==3 -> 1.0, else 0.
    // Built with fma so the ds_load result is unconditionally consumed: no
    // EXEC-predicated loads / saveexec branches in the hot loop.
    float m3 = (lm < 3)  ? 1.0f : 0.0f;
    float a3 = (lm == 3) ? 1.0f : 0.0f;
    v8f clo = {0.f,0.f,0.f,0.f,0.f,0.f,0.f,0.f};   // bins 0..15
    v8f chi = {0.f,0.f,0.f,0.f,0.f,0.f,0.f,0.f};   // bins 16..31
    int baseA = wv * 128;
    for (int s = 0; s < 32; ++s) {
        int a0 = baseA + s * 4 + hh * 2;           // this half's K atoms
        int q0 = q[a0], q1 = q[a0 + 1];
        float p0 = sp[a0 * 3 + chc];
        float p1 = sp[(a0 + 1) * 3 + chc];
        v2f alo, ahi, bm;
        alo.x = (q0 == lm)      ? 1.0f : 0.0f;
        alo.y = (q1 == lm)      ? 1.0f : 0.0f;
        ahi.x = (q0 == lm + 16) ? 1.0f : 0.0f;
        ahi.y = (q1 == lm + 16) ? 1.0f : 0.0f;
        bm.x = fmaf(p0, m3, a3);
        bm.y = fmaf(p1, m3, a3);
        clo = __builtin_amdgcn_wmma_f32_16x16x4_f32(false, alo, false, bm,
                                                    (short)0, clo, false, false);
        chi = __builtin_amdgcn_wmma_f32_16x16x4_f32(false, ahi, false, bm,
                                                    (short)0, chi, false, false);
    }

    // ---- merge 8 waves' partials into LDS bins ----
    if (lm < 4) {
        #pragma unroll
        for (int r = 0; r < 8; ++r) {
            int mlo = r + hh * 8;
            atomicAdd(&bins[mlo * 4 + lm],        clo[r]);
            atomicAdd(&bins[(mlo + 16) * 4 + lm], chi[r]);
        }
    }
    __syncthreads();

    // ---- finalize: disjoint rows [offb, offb+nfr) -> plain global stores ----
    if (t < 128) {
        int bin = t >> 2, c = t & 3;
        if (bin < nfr) {
            int row   = offb + bin;
            float cnt = bins[bin * 4 + 3];
            if (c < 3)
                out[(size_t)row * 3 + c] = bins[bin * 4 + c] / fmaxf(cnt, 1.0f);
            else
                out[(size_t)BG * FG * 3 + row] = cnt;
        }
    }
}

// ---------------------------------------------------------------------------
extern "C" void kernel_launch(void* const* d_in, const int* in_sizes, int n_in,
                              void* d_out, int out_size, void* d_ws, size_t ws_size,
                              hipStream_t stream) {
    (void)in_sizes; (void)n_in; (void)out_size; (void)ws_size;
    const float* pos  = (const float*)d_in[0];
    const int*   frag = (const int*)d_in[1];
    // d_in[2] batch_idx == i / AG by construction; block index supplies it.
    const int*   ent  = (const int*)d_in[3];
    const int*   msk  = (const int*)d_in[4];   // bool mask passed as int32
    float* out = (float*)d_out;

    int* nfrag = (int*)d_ws;          // [BG]
    int* offs  = nfrag + BG;          // [BG]   (64 KB total scratch)

    fmax_kernel<<<BG,   256, 0, stream>>>(frag, nfrag);
    scan_kernel<<<1,    256, 0, stream>>>(nfrag, offs);
    zero_kernel<<<1024, 256, 0, stream>>>(out);
    com_kernel <<<BG,   256, 0, stream>>>(pos, frag, ent, msk, nfrag, offs, out);
}